// AttentionDecoder_6279242187249
// MI455X (gfx1250) — compile-verified
//
#include <hip/hip_runtime.h>
#include <math.h>

typedef __attribute__((ext_vector_type(2))) float v2f;
typedef __attribute__((ext_vector_type(8))) float v8f;

#define BQ    32
#define SDEC  128
#define SENC  128
#define VOCAB 32000
#define EMB   200
#define HID   128
#define G3H   384   // 3*HID

// ---------------------------------------------------------------------------
// 1) x_gates[b,s,g] = dot(emb[seq[b,s]], w_ih[g]) + b_ih[g]
// ---------------------------------------------------------------------------
__global__ void embed_gates_kernel(const int* __restrict__ seq,
                                   const float* __restrict__ emb,
                                   const float* __restrict__ w_ih,
                                   const float* __restrict__ b_ih,
                                   float* __restrict__ x_gates) {
    int idx = blockIdx.x * blockDim.x + threadIdx.x;   // [0, 4096*384)
    if (idx >= BQ * SDEC * G3H) return;
    int row = idx / G3H;     // b*SDEC + s
    int g   = idx % G3H;
    int tok = seq[row];
    const float4* e4 = reinterpret_cast<const float4*>(emb + (size_t)tok * EMB);
    const float4* w4 = reinterpret_cast<const float4*>(w_ih + (size_t)g * EMB);
    float acc = b_ih[g];
    #pragma unroll 5
    for (int e = 0; e < EMB / 4; ++e) {
        float4 a = e4[e];
        float4 w = w4[e];
        acc += a.x * w.x + a.y * w.y + a.z * w.z + a.w * w.w;
    }
    x_gates[idx] = acc;
}

// ---------------------------------------------------------------------------
// 2) GRU scan: single persistent workgroup, h and gate projections in LDS.
// ---------------------------------------------------------------------------
__global__ void gru_kernel(const float* __restrict__ xg,
                           const float* __restrict__ h0,
                           const float* __restrict__ w_hh,
                           const float* __restrict__ b_hh,
                           float* __restrict__ gru_out,
                           float* __restrict__ state_out) {
    __shared__ float h_lds[BQ][HID];     // 16 KB
    __shared__ float hg_lds[BQ][G3H];    // 48 KB
    const int tid = threadIdx.x;         // 0..1023

    for (int i = tid; i < BQ * HID; i += 1024)
        h_lds[i / HID][i % HID] = h0[i];
    __syncthreads();

    for (int t = 0; t < SDEC; ++t) {
        // Phase A: hg = h_prev @ w_hh^T + b_hh   (32 x 384, dot length 128)
        for (int j = tid; j < BQ * G3H; j += 1024) {
            int b = j / G3H, g = j % G3H;
            const float4* w4 = reinterpret_cast<const float4*>(w_hh + (size_t)g * HID);
            const float4* h4 = reinterpret_cast<const float4*>(&h_lds[b][0]);
            float acc = b_hh[g];
            #pragma unroll 8
            for (int k = 0; k < HID / 4; ++k) {
                float4 hv = h4[k];
                float4 wv = w4[k];
                acc += hv.x * wv.x + hv.y * wv.y + hv.z * wv.z + hv.w * wv.w;
            }
            hg_lds[b][g] = acc;
        }
        __syncthreads();

        // Phase B: gate activations + state update (torch gate order r,z,n)
        for (int idx = tid; idx < BQ * HID; idx += 1024) {
            int b = idx / HID, hh = idx % HID;
            const float* xrow = xg + ((size_t)(b * SDEC + t)) * G3H;
            float xr = xrow[hh];
            float xz = xrow[HID + hh];
            float xn = xrow[2 * HID + hh];
            float hr = hg_lds[b][hh];
            float hz = hg_lds[b][HID + hh];
            float hn = hg_lds[b][2 * HID + hh];
            float r = 1.0f / (1.0f + __expf(-(xr + hr)));
            float z = 1.0f / (1.0f + __expf(-(xz + hz)));
            float n = tanhf(xn + r * hn);
            float hprev = h_lds[b][hh];
            float hnew  = (1.0f - z) * n + z * hprev;
            h_lds[b][hh] = hnew;   // only this thread touches (b,hh) in phase B
            gru_out[((size_t)(b * SDEC + t)) * HID + hh] = hnew;
        }
        __syncthreads();
    }

    for (int i = tid; i < BQ * HID; i += 1024)
        state_out[i] = h_lds[i / HID][i % HID];
}

// ---------------------------------------------------------------------------
// 3) s[b,e,d] = dot(hs[b,e,:], gru_out[b,d,:]); softmax over e (encoder axis)
// ---------------------------------------------------------------------------
__global__ void attn_softmax_kernel(const float* __restrict__ hs,
                                    const float* __restrict__ gru_out,
                                    float* __restrict__ att) {
    int bd = blockIdx.x;                 // b*SDEC + d
    int b = bd / SDEC, d = bd % SDEC;
    int e = threadIdx.x;                 // 0..127
    __shared__ float red[SENC];

    const float4* hs4 = reinterpret_cast<const float4*>(hs + ((size_t)(b * SENC + e)) * HID);
    const float4* o4  = reinterpret_cast<const float4*>(gru_out + ((size_t)(b * SDEC + d)) * HID);
    float acc = 0.0f;
    #pragma unroll 8
    for (int k = 0; k < HID / 4; ++k) {
        float4 a = hs4[k];
        float4 o = o4[k];
        acc += a.x * o.x + a.y * o.y + a.z * o.z + a.w * o.w;
    }

    red[e] = acc;
    __syncthreads();
    for (int s = 64; s > 0; s >>= 1) {
        if (e < s) red[e] = fmaxf(red[e], red[e + s]);
        __syncthreads();
    }
    float m = red[0];
    __syncthreads();
    float ex = __expf(acc - m);
    red[e] = ex;
    __syncthreads();
    for (int s = 64; s > 0; s >>= 1) {
        if (e < s) red[e] += red[e + s];
        __syncthreads();
    }
    float inv = 1.0f / red[0];
    att[((size_t)b * SENC + e) * SDEC + d] = ex * inv;
}

// ---------------------------------------------------------------------------
// 4) context c[b,d,h] = sum_e att[b,e,d] * hs[b,e,h]; build cat = [out | c]
// ---------------------------------------------------------------------------
__global__ void context_cat_kernel(const float* __restrict__ att,
                                   const float* __restrict__ hs,
                                   const float* __restrict__ gru_out,
                                   float* __restrict__ cat) {
    int bd = blockIdx.x;
    int b = bd / SDEC, d = bd % SDEC;
    int hcol = threadIdx.x;              // 0..127
    const float* attb = att + (size_t)b * SENC * SDEC;
    const float* hsb  = hs  + (size_t)b * SENC * HID;
    float acc = 0.0f;
    #pragma unroll 4
    for (int e = 0; e < SENC; ++e)
        acc += attb[e * SDEC + d] * hsb[e * HID + hcol];   // attb scalar-broadcast per e
    float o = gru_out[((size_t)bd) * HID + hcol];
    float* crow = cat + ((size_t)bd) * (2 * HID);
    crow[hcol]       = o;
    crow[HID + hcol] = acc;
}

// ---------------------------------------------------------------------------
// 5) logits = cat[4096,256] @ w_out^T[256,32000] + b_out via V_WMMA_F32_16X16X4_F32
//    Register-blocked: block = 256 threads (8 waves), 64-row A panel in LDS.
//    Each wave owns one 16-col N tile and FOUR 16-row M tiles (4 x v8f acc).
//    Hot loop: 1 global b64 (B frag) feeds 4 WMMAs on independent accumulators
//    -> no D->A/B RAW hazards, 4x less w_out L2 traffic than 1-M-tile version.
//    A frag (16x4 f32): lane<16 -> A[lane][k..k+1]; lane>=16 -> A[lane-16][k+2..k+3]
//    B frag (4x16):     v0/v1 = rows {k+kb, k+kb+1}, cols = lane&15
//    D:                 VGPR i -> M = i + (lane<16 ? 0 : 8), N = lane&15
// ---------------------------------------------------------------------------
__global__ void logits_wmma_kernel(const float* __restrict__ cat,
                                   const float* __restrict__ w_out,
                                   const float* __restrict__ b_out,
                                   float* __restrict__ logits) {
    __shared__ float a_lds[64 * 256];    // 64 KB: 64 rows of cat
    const int m0   = blockIdx.y * 64;
    const int wave = threadIdx.x >> 5;
    const int lane = threadIdx.x & 31;
    const int n0   = (blockIdx.x * 8 + wave) * 16;

    // Stage 64x256 A panel (flat copy of 16384 consecutive floats, float4 wide)
    {
        const float4* src = reinterpret_cast<const float4*>(cat + (size_t)m0 * 256);
        float4*       dst = reinterpret_cast<float4*>(a_lds);
        #pragma unroll
        for (int i = 0; i < 16; ++i)
            dst[threadIdx.x + i * 256] = src[threadIdx.x + i * 256];
    }
    __syncthreads();

    const int r  = lane & 15;
    const int kb = (lane < 16) ? 0 : 2;
    const float* abase = a_lds + r * 256;                   // M tile stride = 16*256
    const float* brow  = w_out + ((size_t)(n0 + r)) * 256;  // B[k][n] = w_out[n][k]

    v8f c0 = {}, c1 = {}, c2 = {}, c3 = {};
    #pragma unroll 4
    for (int k = 0; k < 256; k += 4) {
        v2f b  = *reinterpret_cast<const v2f*>(brow + k + kb);
        v2f a0 = *reinterpret_cast<const v2f*>(abase + 0 * 4096 + k + kb);
        v2f a1 = *reinterpret_cast<const v2f*>(abase + 1 * 4096 + k + kb);
        v2f a2 = *reinterpret_cast<const v2f*>(abase + 2 * 4096 + k + kb);
        v2f a3 = *reinterpret_cast<const v2f*>(abase + 3 * 4096 + k + kb);
        c0 = __builtin_amdgcn_wmma_f32_16x16x4_f32(false, a0, false, b, (short)0, c0, false, false);
        c1 = __builtin_amdgcn_wmma_f32_16x16x4_f32(false, a1, false, b, (short)0, c1, false, false);
        c2 = __builtin_amdgcn_wmma_f32_16x16x4_f32(false, a2, false, b, (short)0, c2, false, false);
        c3 = __builtin_amdgcn_wmma_f32_16x16x4_f32(false, a3, false, b, (short)0, c3, false, false);
    }

    const int ncol = n0 + r;
    const float bias = b_out[ncol];
    const int mbase = m0 + ((lane < 16) ? 0 : 8);
    v8f acc[4] = {c0, c1, c2, c3};
    #pragma unroll
    for (int mt = 0; mt < 4; ++mt) {
        const int mrow = mbase + mt * 16;
        #pragma unroll
        for (int i = 0; i < 8; ++i)
            logits[((size_t)(mrow + i)) * VOCAB + ncol] = acc[mt][i] + bias;
    }
}

// ---------------------------------------------------------------------------
extern "C" void kernel_launch(void* const* d_in, const int* in_sizes, int n_in,
                              void* d_out, int out_size, void* d_ws, size_t ws_size,
                              hipStream_t stream) {
    (void)in_sizes; (void)n_in; (void)out_size; (void)ws_size;

    const int*   seq   = (const int*)  d_in[0];
    const float* hs    = (const float*)d_in[1];
    const float* h0    = (const float*)d_in[2];
    const float* emb   = (const float*)d_in[3];
    const float* w_ih  = (const float*)d_in[4];
    const float* w_hh  = (const float*)d_in[5];
    const float* b_ih  = (const float*)d_in[6];
    const float* b_hh  = (const float*)d_in[7];
    const float* w_out = (const float*)d_in[8];
    const float* b_out = (const float*)d_in[9];

    // Output layout: logits [4096,32000] | state [1,32,128] | att [32,128,128]
    float* logits = (float*)d_out;
    float* state  = logits + (size_t)BQ * SDEC * VOCAB;
    float* att    = state  + (size_t)BQ * HID;

    // Workspace layout (floats)
    float* x_gates = (float*)d_ws;                          // 4096*384
    float* gru_out = x_gates + (size_t)BQ * SDEC * G3H;     // 4096*128
    float* cat     = gru_out + (size_t)BQ * SDEC * HID;     // 4096*256

    embed_gates_kernel<<<(BQ * SDEC * G3H) / 256, 256, 0, stream>>>(
        seq, emb, w_ih, b_ih, x_gates);

    gru_kernel<<<1, 1024, 0, stream>>>(
        x_gates, h0, w_hh, b_hh, gru_out, state);

    attn_softmax_kernel<<<BQ * SDEC, SENC, 0, stream>>>(
        hs, gru_out, att);

    context_cat_kernel<<<BQ * SDEC, HID, 0, stream>>>(
        att, hs, gru_out, cat);

    dim3 g5(VOCAB / (16 * 8), (BQ * SDEC) / 64);  // 250 x 64 blocks
    logits_wmma_kernel<<<g5, 256, 0, stream>>>(
        cat, w_out, b_out, logits);
}